// TaskAlignedAssigner_66511863546261
// MI455X (gfx1250) — compile-verified
//
#include <hip/hip_runtime.h>
#include <hip/hip_bf16.h>
#include <cstdint>
#include <cstddef>

#define BS   32
#define NA   8400
#define NG   64
#define NC   80
#define KTOP 13
#define EPSF 1e-9f

typedef float v2f __attribute__((ext_vector_type(2)));
typedef float v8f __attribute__((ext_vector_type(8)));

__device__ __forceinline__ float iou_box(const float4 g, const float4 p) {
    float lx = fmaxf(g.x, p.x), ly = fmaxf(g.y, p.y);
    float rx = fminf(g.z, p.z), ry = fminf(g.w, p.w);
    float inter = fmaxf(rx - lx, 0.f) * fmaxf(ry - ly, 0.f);
    float a1 = fmaxf(g.z - g.x, 0.f) * fmaxf(g.w - g.y, 0.f);
    float a2 = fmaxf(p.z - p.x, 0.f) * fmaxf(p.w - p.y, 0.f);
    return inter / (a1 + a2 - inter + EPSF);
}

// ---------------------------------------------------------------------------
// Kernel 1: bbox_scores[b,j,a] = pd_scores[b,a,gl[b,j]] computed as
//           P(A x C) @ onehot(labels)(C x 16) via V_WMMA_F32_16X16X4_F32.
// Tile: M=16 anchors, N=16 gts, K=4 classes, 20 k-steps over C=80.
// ---------------------------------------------------------------------------
__global__ __launch_bounds__(256) void bbox_scores_wmma(
        const float* __restrict__ pd_scores,
        const int*   __restrict__ gt_labels,
        float*       __restrict__ scores_out) {
    const int wave = (int)((blockIdx.x * blockDim.x + threadIdx.x) >> 5);
    const int lane = (int)(threadIdx.x & 31);
    // tiles: b(32) * ag(525) * gg(4)
    const int gg = wave & 3;
    const int ag = (wave >> 2) % 525;
    const int b  = wave / (525 * 4);

    const int half = lane >> 4;      // 0: lanes 0-15, 1: lanes 16-31
    const int ml   = lane & 15;

    const int lab = gt_labels[b * NG + gg * 16 + ml];   // one-hot column for this lane
    const float* Abase = pd_scores + ((size_t)b * NA + (size_t)(ag * 16 + ml)) * NC;

    v8f c = {};
    for (int k0 = 0; k0 < NC; k0 += 4) {
        // A 16x4 f32: lane = M; VGPR0/1 = K,K+1 (lanes<16) or K+2,K+3 (lanes>=16)
        const int kk = k0 + 2 * half;
        v2f a;
        a.x = Abase[kk];
        a.y = Abase[kk + 1];
        // B 4x16 f32 one-hot: VGPR0 rows k0(+half), VGPR1 rows k0+2(+half), col = ml
        v2f bm;
        bm.x = (lab == (k0 + half))     ? 1.f : 0.f;
        bm.y = (lab == (k0 + 2 + half)) ? 1.f : 0.f;
        c = __builtin_amdgcn_wmma_f32_16x16x4_f32(false, a, false, bm,
                                                  (short)0, c, false, false);
    }
    // C/D 16x16 f32 layout: VGPR v -> M = v + 8*half, N = ml
    float* out = scores_out + (size_t)b * NG * NA
               + (size_t)(gg * 16 + ml) * NA
               + (size_t)(ag * 16 + half * 8);
#pragma unroll
    for (int v = 0; v < 8; ++v) out[v] = c[v];
}

// ---------------------------------------------------------------------------
// Kernel 2: per (b,j): align_metric (in-place over scores_ws), masked metric
// row in LDS, iterative top-13 selection, scatter gt-membership bits.
// ---------------------------------------------------------------------------
__global__ __launch_bounds__(256) void metrics_topk(
        const float* __restrict__ pd_bboxes,
        const float* __restrict__ anc,
        const float* __restrict__ gt_bboxes,
        const float* __restrict__ mask_gt,
        float*       __restrict__ scores_align,      // in: bbox_scores, out: align_metric
        unsigned long long* __restrict__ mask_bits) {
    __shared__ float met[NA];
    __shared__ unsigned long long red[256];

    const int j = (int)blockIdx.x;
    const int b = (int)blockIdx.y;
    const int tid = (int)threadIdx.x;

    const float4 g = ((const float4*)gt_bboxes)[b * NG + j];
    const size_t row = ((size_t)b * NG + j) * NA;
    const float4* pbox = (const float4*)pd_bboxes + (size_t)b * NA;

    for (int a = tid; a < NA; a += 256) {
        if (a + 1024 < NA) __builtin_prefetch(&scores_align[row + a + 1024], 0, 0);
        const float v = iou_box(g, pbox[a]);
        const float s = scores_align[row + a];
        const float t = v * v;
        const float al = s * (t * t * t);            // score^1 * iou^6
        scores_align[row + a] = al;                  // raw align_metric (pre-mask)
        const float2 ap = ((const float2*)anc)[a];
        const bool ing = (ap.x - g.x > EPSF) && (ap.y - g.y > EPSF) &&
                         (g.z - ap.x > EPSF) && (g.w - ap.y > EPSF);
        met[a] = ing ? al : 0.f;
    }
    __syncthreads();

    if (mask_gt[b * NG + j] == 0.f) return;          // uniform across block

    for (int it = 0; it < KTOP; ++it) {
        // local argmax with top_k tie-break (value desc, index asc)
        unsigned long long best = 0ull;
        for (int a = tid; a < NA; a += 256) {
            unsigned u = __float_as_uint(met[a]);
            u ^= (u & 0x80000000u) ? 0xFFFFFFFFu : 0x80000000u;  // order-preserving map
            const unsigned long long key =
                ((unsigned long long)u << 32) | (unsigned)(NA - a);
            if (key > best) best = key;
        }
        red[tid] = best;
        __syncthreads();
        for (int s = 128; s > 0; s >>= 1) {
            if (tid < s) { if (red[tid + s] > red[tid]) red[tid] = red[tid + s]; }
            __syncthreads();
        }
        if (tid == 0) {
            const unsigned long long key = red[0];
            const int amax = NA - (int)(unsigned)(key & 0xFFFFFFFFull);
            met[amax] = __int_as_float(0xff800000);  // -inf: remove from next rounds
            const float2 ap = ((const float2*)anc)[amax];
            const bool ing = (ap.x - g.x > EPSF) && (ap.y - g.y > EPSF) &&
                             (g.z - ap.x > EPSF) && (g.w - ap.y > EPSF);
            if (ing)  // mask_in_gts applied to the selected top-k index
                atomicOr(&mask_bits[(size_t)b * NA + amax], 1ull << j);
        }
        __syncthreads();
    }
}

// ---------------------------------------------------------------------------
// Kernel 3: resolve anchors assigned to >1 gt (argmax IoU over ALL gts),
// then per-gt maxima pos_align / pos_overlaps via float-as-int atomicMax.
// ---------------------------------------------------------------------------
__global__ __launch_bounds__(256) void resolve_pos(
        const float* __restrict__ pd_bboxes,
        const float* __restrict__ gt_bboxes,
        const float* __restrict__ scores_align,
        unsigned long long* __restrict__ mask_bits,
        float* __restrict__ pos_align,
        float* __restrict__ pos_ov) {
    __shared__ float4 gb[NG];
    const int b = (int)blockIdx.y;
    const int tid = (int)threadIdx.x;
    if (tid < NG) gb[tid] = ((const float4*)gt_bboxes)[b * NG + tid];
    __syncthreads();

    const int a = (int)blockIdx.x * 256 + tid;
    if (a >= NA) return;
    const size_t ai = (size_t)b * NA + a;
    unsigned long long bits = mask_bits[ai];
    if (!bits) return;

    const float4 p = ((const float4*)pd_bboxes)[ai];
    if (__popcll(bits) > 1) {
        float bestv = -1.f; int bj = 0;
        for (int jj = 0; jj < NG; ++jj) {
            const float v = iou_box(gb[jj], p);
            if (v > bestv) { bestv = v; bj = jj; }   // first max wins (argmax semantics)
        }
        bits = 1ull << bj;
        mask_bits[ai] = bits;
    }
    const int j = __ffsll((unsigned long long)bits) - 1;
    const float al = scores_align[((size_t)b * NG + j) * NA + a];
    const float ov = iou_box(gb[j], p);
    atomicMax((int*)&pos_align[b * NG + j], __float_as_int(al));  // values >= 0
    atomicMax((int*)&pos_ov[b * NG + j],    __float_as_int(ov));
}

// ---------------------------------------------------------------------------
// Kernel 4: per-anchor labels / bboxes / fg / norm.
// ---------------------------------------------------------------------------
__global__ __launch_bounds__(256) void write_anchor_outputs(
        const int*   __restrict__ gt_labels,
        const float* __restrict__ gt_bboxes,
        const float* __restrict__ scores_align,
        const unsigned long long* __restrict__ mask_bits,
        const float* __restrict__ pos_align,
        const float* __restrict__ pos_ov,
        float* __restrict__ out_labels,
        float* __restrict__ out_bboxes,
        float* __restrict__ out_fg,
        float* __restrict__ norm_ws,
        int*   __restrict__ labsel_ws) {
    const int b = (int)blockIdx.y;
    const int a = (int)blockIdx.x * 256 + (int)threadIdx.x;
    if (a >= NA) return;
    const size_t ai = (size_t)b * NA + a;
    const unsigned long long bits = mask_bits[ai];
    const bool fg = bits != 0ull;
    const int tgt = fg ? (__ffsll((unsigned long long)bits) - 1) : 0;

    int lab = gt_labels[b * NG + tgt];
    lab = lab < 0 ? 0 : lab;
    out_labels[ai] = (float)lab;
    ((float4*)out_bboxes)[ai] = ((const float4*)gt_bboxes)[b * NG + tgt];
    out_fg[ai] = fg ? 1.f : 0.f;

    float norm = 0.f;
    if (fg) {
        const float al = scores_align[((size_t)b * NG + tgt) * NA + a];
        norm = al * pos_ov[b * NG + tgt] / (pos_align[b * NG + tgt] + EPSF);
    }
    norm_ws[ai] = norm;
    labsel_ws[ai] = fg ? lab : -1;
}

// ---------------------------------------------------------------------------
// Kernel 5: coalesced (b, A, 80) one-hot * norm score write.
// ---------------------------------------------------------------------------
__global__ __launch_bounds__(256) void write_scores(
        const float* __restrict__ norm_ws,
        const int*   __restrict__ labsel_ws,
        float*       __restrict__ out_scores) {
    const size_t i = (size_t)blockIdx.x * 256 + threadIdx.x;
    if (i >= (size_t)BS * NA * NC) return;
    const size_t ba = i / NC;
    const int c = (int)(i - ba * NC);
    out_scores[i] = (c == labsel_ws[ba]) ? norm_ws[ba] : 0.f;
}

// ---------------------------------------------------------------------------
extern "C" void kernel_launch(void* const* d_in, const int* in_sizes, int n_in,
                              void* d_out, int out_size, void* d_ws, size_t ws_size,
                              hipStream_t stream) {
    const float* pd_scores = (const float*)d_in[0];
    const float* pd_bboxes = (const float*)d_in[1];
    const float* anc       = (const float*)d_in[2];
    const int*   gt_labels = (const int*)d_in[3];
    const float* gt_bboxes = (const float*)d_in[4];
    const float* mask_gt   = (const float*)d_in[5];

    char* ws = (char*)d_ws;
    float* scores_align            = (float*)(ws);                       // 68,812,800 B
    unsigned long long* mask_bits  = (unsigned long long*)(ws + 68812800); // 2,150,400 B
    float* pos_align               = (float*)(ws + 70963200);            // 8,192 B
    float* pos_ov                  = (float*)(ws + 70971392);            // 8,192 B
    float* norm_ws                 = (float*)(ws + 70979584);            // 1,075,200 B
    int*   labsel_ws               = (int*)  (ws + 72054784);            // 1,075,200 B

    float* out = (float*)d_out;
    const size_t L = (size_t)BS * NA;
    float* out_labels = out;            // (b, A)
    float* out_bboxes = out + L;        // (b, A, 4)
    float* out_scores = out + 5 * L;    // (b, A, 80)
    float* out_fg     = out + 85 * L;   // (b, A)

    hipMemsetAsync(mask_bits, 0, L * sizeof(unsigned long long), stream);
    hipMemsetAsync(pos_align, 0, (size_t)BS * NG * sizeof(float), stream);
    hipMemsetAsync(pos_ov,    0, (size_t)BS * NG * sizeof(float), stream);

    // 1) one-hot GEMM via WMMA: 32 * 525 * 4 = 67,200 tiles, 8 waves / block
    bbox_scores_wmma<<<8400, 256, 0, stream>>>(pd_scores, gt_labels, scores_align);

    // 2) align metric + top-13 per (b, gt)
    dim3 gB(NG, BS);
    metrics_topk<<<gB, 256, 0, stream>>>(pd_bboxes, anc, gt_bboxes, mask_gt,
                                         scores_align, mask_bits);

    // 3) multi-assignment resolution + per-gt maxima
    dim3 gA((NA + 255) / 256, BS);
    resolve_pos<<<gA, 256, 0, stream>>>(pd_bboxes, gt_bboxes, scores_align,
                                        mask_bits, pos_align, pos_ov);

    // 4) per-anchor outputs
    write_anchor_outputs<<<gA, 256, 0, stream>>>(gt_labels, gt_bboxes, scores_align,
                                                 mask_bits, pos_align, pos_ov,
                                                 out_labels, out_bboxes, out_fg,
                                                 norm_ws, labsel_ws);

    // 5) coalesced score tensor
    const size_t nsc = (size_t)BS * NA * NC;
    write_scores<<<(unsigned)((nsc + 255) / 256), 256, 0, stream>>>(norm_ws, labsel_ws,
                                                                    out_scores);
}